// Whiten2d_29016799052491
// MI455X (gfx1250) — compile-verified
//
#include <hip/hip_runtime.h>
#include <hip/hip_bf16.h>

// Whiten2d (IterNorm-style ZCA whitening) for MI455X / gfx1250.
// Memory-bound (~384 MiB HBM traffic, ~9 GFLOP fp32) -> stay fp32, use
// V_WMMA_F32_16X16X4_F32 for all GEMM stages. C=64 == 4x4 grid of 16x16 tiles.
// Cov SYRK uses GLOBAL_LOAD_ASYNC_TO_LDS_B128 double-buffering (ASYNCcnt)
// so the dominant 128 MiB read overlaps with WMMA compute.

#define NS   32      // batch
#define CC   64      // channels
#define HWS  16384   // H*W
#define EPSV 1e-5f
#define TIT  5

typedef float v2f __attribute__((ext_vector_type(2)));
typedef float v8f __attribute__((ext_vector_type(8)));

static __device__ __forceinline__ v8f wmma_f32(v2f a, v2f b, v8f c) {
  // V_WMMA_F32_16X16X4_F32: D = A(16x4) * B(4x16) + C(16x16), all fp32
  return __builtin_amdgcn_wmma_f32_16x16x4_f32(false, a, false, b, (short)0, c,
                                               false, false);
}

// ---------------------------------------------------------------- mean ----
__global__ void mean_kernel(const float* __restrict__ x, float* __restrict__ mean) {
  __shared__ float red[256];
  const int t = threadIdx.x;
  const float4* xv = (const float4*)(x + (size_t)blockIdx.x * HWS);
  float s = 0.f;
#pragma unroll
  for (int i = 0; i < 16; ++i) {
    float4 v = xv[t + i * 256];
    s += (v.x + v.y) + (v.z + v.w);
  }
  red[t] = s;
  __syncthreads();
  for (int off = 128; off > 0; off >>= 1) {
    if (t < off) red[t] += red[t + off];
    __syncthreads();
  }
  if (t == 0) mean[blockIdx.x] = red[0] * (1.0f / HWS);
}

// ----------------------------------------------------------------- cov ----
// Raw Gram G = x x^T via WMMA, then cov = G/S - mu mu^T + eps*I.
// One block per sample, 8 waves; async mem->LDS double buffering.
#define BK 128
#define LP 132   // BK + 4: 16B-aligned B128 async stores, conflict-free v2f loads

static __device__ __forceinline__ void async_copy_slab(const float* __restrict__ xn,
                                                       unsigned ldsbase, int kb, int t) {
  // 64 rows x 128 cols fp32 = 2048 x 16B chunks; 8 per thread, coalesced in s.
#pragma unroll
  for (int it = 0; it < 8; ++it) {
    int idx = t + it * 256;
    int c   = idx >> 5;
    int k4  = (idx & 31) << 2;
    unsigned lds  = ldsbase + (unsigned)((c * LP + k4) << 2);
    unsigned voff = (unsigned)((c * HWS + kb + k4) << 2);
    asm volatile("global_load_async_to_lds_b128 %0, %1, %2 offset:0"
                 :: "v"(lds), "v"(voff), "s"(xn)
                 : "memory");
  }
}

__global__ void cov_kernel(const float* __restrict__ x, const float* __restrict__ mean,
                           float* __restrict__ cov) {
  extern __shared__ float lbuf[];         // 2 * CC*LP floats (double buffer)
  __shared__ float mu[CC];
  const int n = blockIdx.x;
  const int t = threadIdx.x;
  const int wave = t >> 5, lane = t & 31;
  if (t < CC) mu[t] = mean[n * CC + t];

  // wave owns tiles (i, j0) and (i, j1) of the 4x4 tile grid (shared A row)
  const int i  = wave >> 1;
  const int j0 = (wave & 1) << 1;
  const int j1 = j0 + 1;
  const int m  = lane & 15;               // M (A) / N (B) index
  const int kh = (lane >> 4) << 1;        // K half offset: 0 or 2

  v8f acc0 = {}; v8f acc1 = {};
  const float* xn = x + (size_t)n * CC * HWS;
  const unsigned ldsbase = (unsigned)(size_t)(void*)lbuf;
  const unsigned bufstep = (unsigned)(CC * LP) << 2;

  async_copy_slab(xn, ldsbase, 0, t);     // prologue: fill buffer 0

  for (int kb = 0; kb < HWS; kb += BK) {
    const int cur = (kb >> 7) & 1;
    // own async loads landed; barrier => all waves' data visible AND everyone
    // is done reading the buffer we are about to overwrite next.
    asm volatile("s_wait_asynccnt 0x0" ::: "memory");
    __syncthreads();
    if (kb + BK < HWS)
      async_copy_slab(xn, ldsbase + (unsigned)(cur ^ 1) * bufstep, kb + BK, t);

    const float* lx = lbuf + cur * (CC * LP);
#pragma unroll 4
    for (int k0 = 0; k0 < BK; k0 += 4) {
      int col = k0 + kh;
      v2f a  = *(const v2f*)&lx[(16 * i  + m) * LP + col];
      v2f b0 = *(const v2f*)&lx[(16 * j0 + m) * LP + col];  // B = x^T: same layout
      v2f b1 = *(const v2f*)&lx[(16 * j1 + m) * LP + col];
      acc0 = wmma_f32(a, b0, acc0);
      acc1 = wmma_f32(a, b1, acc1);
    }
  }

  const int rb = 16 * i + ((lane >> 4) << 3);   // C/D: M in 0..7 / 8..15 halves
  const int c0 = 16 * j0 + m;
  const int c1 = 16 * j1 + m;
  const float mu0 = mu[c0], mu1 = mu[c1];
  float* covn = cov + (size_t)n * CC * CC;
#pragma unroll
  for (int r = 0; r < 8; ++r) {
    int ci = rb + r;
    float mc = mu[ci];
    covn[ci * CC + c0] = acc0[r] * (1.0f / HWS) - mc * mu0 + (ci == c0 ? EPSV : 0.0f);
    covn[ci * CC + c1] = acc1[r] * (1.0f / HWS) - mc * mu1 + (ci == c1 ? EPSV : 0.0f);
  }
}

// --------------------------------------------------------- Newton-Schulz ----
#define MP 66   // 64 + 2: even pitch for aligned v2f A-fragment loads

static __device__ __forceinline__ void mm64_frag(const float* __restrict__ A,
                                                 const float* __restrict__ B,
                                                 int i, int j0, int j1, int m, int kh,
                                                 v8f& a0, v8f& a1) {
#pragma unroll
  for (int k0 = 0; k0 < CC; k0 += 4) {
    int col = k0 + kh;
    v2f a = *(const v2f*)&A[(16 * i + m) * MP + col];
    v2f b0, b1;
    b0.x = B[col * MP + 16 * j0 + m];
    b0.y = B[(col + 1) * MP + 16 * j0 + m];
    b1.x = B[col * MP + 16 * j1 + m];
    b1.y = B[(col + 1) * MP + 16 * j1 + m];
    a0 = wmma_f32(a, b0, a0);
    a1 = wmma_f32(a, b1, a1);
  }
}

static __device__ __forceinline__ void store_tiles(float* __restrict__ O,
                                                   int i, int j0, int j1, int m, int rb16,
                                                   const v8f& a0, const v8f& a1) {
#pragma unroll
  for (int r = 0; r < 8; ++r) {
    int ri = 16 * i + rb16 + r;
    O[ri * MP + 16 * j0 + m] = a0[r];
    O[ri * MP + 16 * j1 + m] = a1[r];
  }
}

__global__ void newton_kernel(const float* __restrict__ cov, float* __restrict__ wm) {
  __shared__ float sC[CC * MP];   // trace-normalized cov
  __shared__ float sP[CC * MP];   // P
  __shared__ float sT[CC * MP];   // temp: P2 then P3 (register-staged overwrite)
  __shared__ float sS[1];
  const int n = blockIdx.x;
  const int t = threadIdx.x;
  const int wave = t >> 5, lane = t & 31;
  const int i  = wave >> 1;
  const int j0 = (wave & 1) << 1;
  const int j1 = j0 + 1;
  const int m  = lane & 15;
  const int kh = (lane >> 4) << 1;
  const int rb16 = (lane >> 4) << 3;

  const float* covn = cov + (size_t)n * CC * CC;
#pragma unroll
  for (int it = 0; it < 16; ++it) {
    int idx = t + it * 256;
    int r = idx >> 6, c = idx & 63;
    sC[r * MP + c] = covn[idx];
    sP[r * MP + c] = (r == c) ? 1.0f : 0.0f;
  }
  __syncthreads();
  if (t == 0) {
    float tr = 0.f;
    for (int c = 0; c < CC; ++c) tr += sC[c * MP + c];
    sS[0] = 1.0f / tr;
  }
  __syncthreads();
  const float rTr = sS[0];
#pragma unroll
  for (int it = 0; it < 16; ++it) {
    int idx = t + it * 256;
    sC[(idx >> 6) * MP + (idx & 63)] *= rTr;
  }
  __syncthreads();

  for (int iter = 0; iter < TIT; ++iter) {
    // T = P @ P
    v8f a0 = {}, a1 = {};
    mm64_frag(sP, sP, i, j0, j1, m, kh, a0, a1);
    store_tiles(sT, i, j0, j1, m, rb16, a0, a1);
    __syncthreads();
    // T <- T @ P   (P3; stage in regs, barrier, overwrite)
    a0 = (v8f){}; a1 = (v8f){};
    mm64_frag(sT, sP, i, j0, j1, m, kh, a0, a1);
    __syncthreads();
    store_tiles(sT, i, j0, j1, m, rb16, a0, a1);
    __syncthreads();
    // P <- 1.5*P - 0.5*(T @ covN)
    a0 = (v8f){}; a1 = (v8f){};
    mm64_frag(sT, sC, i, j0, j1, m, kh, a0, a1);
#pragma unroll
    for (int r = 0; r < 8; ++r) {
      int ri = 16 * i + rb16 + r;
      int cA = 16 * j0 + m, cB = 16 * j1 + m;
      sP[ri * MP + cA] = 1.5f * sP[ri * MP + cA] - 0.5f * a0[r];
      sP[ri * MP + cB] = 1.5f * sP[ri * MP + cB] - 0.5f * a1[r];
    }
    __syncthreads();
  }

  const float srt = sqrtf(rTr);
  float* wmn = wm + (size_t)n * CC * CC;
#pragma unroll
  for (int it = 0; it < 16; ++it) {
    int idx = t + it * 256;
    wmn[idx] = sP[(idx >> 6) * MP + (idx & 63)] * srt;
  }
}

// --------------------------------------------------------------- apply ----
// out = (wm @ xc) * weight + bias ; grid (N, 16 s-chunks of 1024)
__global__ void apply_kernel(const float* __restrict__ x, const float* __restrict__ mean,
                             const float* __restrict__ wm, const float* __restrict__ weight,
                             const float* __restrict__ bias, float* __restrict__ out) {
  __shared__ float sW[CC * MP];
  __shared__ float sMu[CC], sGa[CC], sBe[CC];
  const int n = blockIdx.x;
  const int t = threadIdx.x;
  const int wave = t >> 5, lane = t & 31;
  const int m  = lane & 15;
  const int kh = (lane >> 4) << 1;
  const int rb16 = (lane >> 4) << 3;

  const float* wmn = wm + (size_t)n * CC * CC;
#pragma unroll
  for (int it = 0; it < 16; ++it) {
    int idx = t + it * 256;
    sW[(idx >> 6) * MP + (idx & 63)] = wmn[idx];
  }
  if (t < CC) {
    sMu[t] = mean[n * CC + t];
    sGa[t] = weight[t];
    sBe[t] = bias[t];
  }
  __syncthreads();

  const float* xn = x + (size_t)n * CC * HWS;
  float* on = out + (size_t)n * CC * HWS;
  const int chunk = blockIdx.y << 10;

  for (int st = 0; st < 8; ++st) {
    const int sbase = chunk + ((st * 8 + wave) << 4);
    v8f acc0 = {}, acc1 = {}, acc2 = {}, acc3 = {};
#pragma unroll 4
    for (int k0 = 0; k0 < CC; k0 += 4) {
      const int kr = k0 + kh;
      v2f b;  // B = xc[kr..kr+1][sbase+m], centered on the fly
      b.x = xn[(size_t)kr * HWS + sbase + m] - sMu[kr];
      b.y = xn[(size_t)(kr + 1) * HWS + sbase + m] - sMu[kr + 1];
      v2f A0 = *(const v2f*)&sW[(0  + m) * MP + kr];
      v2f A1 = *(const v2f*)&sW[(16 + m) * MP + kr];
      v2f A2 = *(const v2f*)&sW[(32 + m) * MP + kr];
      v2f A3 = *(const v2f*)&sW[(48 + m) * MP + kr];
      acc0 = wmma_f32(A0, b, acc0);
      acc1 = wmma_f32(A1, b, acc1);
      acc2 = wmma_f32(A2, b, acc2);
      acc3 = wmma_f32(A3, b, acc3);
    }
    const int s = sbase + m;
#pragma unroll
    for (int r = 0; r < 8; ++r) {
      int c = rb16 + r;
      on[(size_t)(c)      * HWS + s] = acc0[r] * sGa[c]      + sBe[c];
      on[(size_t)(c + 16) * HWS + s] = acc1[r] * sGa[c + 16] + sBe[c + 16];
      on[(size_t)(c + 32) * HWS + s] = acc2[r] * sGa[c + 32] + sBe[c + 32];
      on[(size_t)(c + 48) * HWS + s] = acc3[r] * sGa[c + 48] + sBe[c + 48];
    }
  }
}

// -------------------------------------------------------------- launch ----
extern "C" void kernel_launch(void* const* d_in, const int* in_sizes, int n_in,
                              void* d_out, int out_size, void* d_ws, size_t ws_size,
                              hipStream_t stream) {
  (void)in_sizes; (void)n_in; (void)out_size; (void)ws_size;
  const float* x      = (const float*)d_in[0];
  const float* weight = (const float*)d_in[1];
  const float* bias   = (const float*)d_in[2];
  float* out = (float*)d_out;

  float* mean = (float*)d_ws;            // NS*CC            = 2048 floats
  float* cov  = mean + 2048;             // NS*CC*CC         = 131072 floats
  float* wm   = cov + NS * CC * CC;      // NS*CC*CC         = 131072 floats

  mean_kernel<<<NS * CC, 256, 0, stream>>>(x, mean);
  const size_t covShm = (size_t)2 * CC * LP * sizeof(float);  // 67,584 B (< 320 KB/WGP)
  cov_kernel<<<NS, 256, covShm, stream>>>(x, mean, cov);
  newton_kernel<<<NS, 256, 0, stream>>>(cov, wm);
  dim3 grid(NS, 16);
  apply_kernel<<<grid, 256, 0, stream>>>(x, mean, wm, weight, bias, out);
}